// Model_1632087573207
// MI455X (gfx1250) — compile-verified
//
#include <hip/hip_runtime.h>
#include <math.h>
#include <stdint.h>

// ---- problem constants (fixed by the reference) ----
#define NN   128
#define CC   64
#define TT   288
#define VV   16
#define SS   3
#define ICN  16
#define TVN  (TT * VV)          // 4608
#define BN_EPS 1e-5f

typedef float v8f __attribute__((ext_vector_type(8)));
typedef float v2f __attribute__((ext_vector_type(2)));

// D = A(16x4 f32) * B(4x16 f32) + C(16x16 f32), full fp32 precision.
__device__ __forceinline__ v8f wmma4(v2f a, v2f b, v8f c) {
  return __builtin_amdgcn_wmma_f32_16x16x4_f32(
      /*neg_a=*/false, a, /*neg_b=*/false, b,
      /*c_mod=*/(short)0, c, /*reuse_a=*/false, /*reuse_b=*/false);
}

__device__ __forceinline__ v8f v8f_zero() {
  v8f r;
#pragma unroll
  for (int i = 0; i < 8; ++i) r[i] = 0.0f;
  return r;
}

// ---- CDNA5 async global->LDS copy (ASYNCcnt-tracked, no VGPR data path) ----
// Copies 64 contiguous bytes; INST_OFFSET is applied to BOTH the global and
// LDS addresses (CDNA5 ISA), so one address pair suffices.
// ldsaddr = low 32 bits of the generic pointer == LDS byte address.
__device__ __forceinline__ void async_cp64B(uint32_t ldsaddr, const float* g) {
  asm volatile(
      "global_load_async_to_lds_b128 %0, %1, off\n\t"
      "global_load_async_to_lds_b128 %0, %1, off offset:16\n\t"
      "global_load_async_to_lds_b128 %0, %1, off offset:32\n\t"
      "global_load_async_to_lds_b128 %0, %1, off offset:48"
      :: "v"(ldsaddr), "v"(g) : "memory");
}
__device__ __forceinline__ void wait_async0() {
  asm volatile("s_wait_asynccnt 0x0" ::: "memory");
}
__device__ __forceinline__ uint32_t lds_addr32(const void* p) {
  return (uint32_t)(uintptr_t)p;      // aperture: addr[31:0] = LDS byte addr
}

// =====================================================================
// Kernel 1: per (n,s) compute Att[n,s] (16x16) = softmax_v(G/4608) + A
//           G = sum_{i,t} fa[i,t,:] outer fb[i,t,:],  f* = W* @ X_t + b*
// =====================================================================
__global__ __launch_bounds__(256)
void attn_kernel(const float* __restrict__ x,
                 const float* __restrict__ A_base, const float* __restrict__ PA,
                 const float* __restrict__ Wa, const float* __restrict__ ba,
                 const float* __restrict__ Wb, const float* __restrict__ bb,
                 float* __restrict__ att_out)
{
  const int s    = blockIdx.x % SS;
  const int n    = blockIdx.x / SS;
  const int tid  = threadIdx.x;
  const int wave = tid >> 5;          // 0..7
  const int lane = tid & 31;
  const int lm   = lane & 15;         // M / N (row/col) lane index
  const int lh   = lane >> 4;         // K half select (0 or 1)

  __shared__ float sW[2][ICN * CC];       // Wa[s], Wb[s]          (8 KB)
  __shared__ float sFa[16 * ICN * VV];    // [t][i][v] chunk       (16 KB)
  __shared__ float sFb[16 * ICN * VV];    //                       (16 KB)
  __shared__ float sG[8][VV * VV];        // per-wave partial G    (8 KB)
  __shared__ float sGsum[VV * VV];        //                       (1 KB)

  for (int e = tid; e < ICN * CC; e += 256) {
    sW[0][e] = Wa[s * ICN * CC + e];
    sW[1][e] = Wb[s * ICN * CC + e];
  }
  __syncthreads();

  // Hoist Wa/Wb A-operands into VGPRs: A[m=i, k=c]; lane holds (c0, c0+1).
  v2f waA[16], wbA[16];
#pragma unroll
  for (int ks = 0; ks < 16; ++ks) {
    const int c0 = 4 * ks + 2 * lh;
    const int ix = lm * CC + c0;
    waA[ks].x = sW[0][ix];  waA[ks].y = sW[0][ix + 1];
    wbA[ks].x = sW[1][ix];  wbA[ks].y = sW[1][ix + 1];
  }
  // Per-accum-VGPR bias (D row for VGPR r is i = r + 8*lh).
  float bAr[8], bBr[8];
#pragma unroll
  for (int r = 0; r < 8; ++r) {
    bAr[r] = ba[s * ICN + r + 8 * lh];
    bBr[r] = bb[s * ICN + r + 8 * lh];
  }

  const float* xn = x + (size_t)n * CC * TVN;
  v8f Gacc = v8f_zero();

  for (int chunk = 0; chunk < TT / 16; ++chunk) {
    const int t0 = chunk * 16;
    // Each wave computes fa,fb for two t's of this 16-t chunk.
#pragma unroll
    for (int tt = 0; tt < 2; ++tt) {
      const int tl = wave * 2 + tt;      // local t  (0..15)
      const int tg = t0 + tl;            // global t
      v8f fa = v8f_zero(), fb = v8f_zero();
#pragma unroll
      for (int ks = 0; ks < 16; ++ks) {
        const int c0 = 4 * ks + 2 * lh;
        const float* p0 = xn + (size_t)c0 * TVN + (size_t)tg * VV + lm;
        v2f bx;                          // B[k=c, n=v] straight from global
        bx.x = p0[0];
        bx.y = p0[TVN];
        fa = wmma4(waA[ks], bx, fa);     // shared B operand for both GEMMs
        fb = wmma4(wbA[ks], bx, fb);
      }
      // D layout: VGPR r -> (i = r+8*lh, v = lm). Store as k = tl*16 + i.
#pragma unroll
      for (int r = 0; r < 8; ++r) {
        const int k = tl * ICN + r + 8 * lh;
        sFa[k * VV + lm] = fa[r] + bAr[r];
        sFb[k * VV + lm] = fb[r] + bBr[r];
      }
    }
    __syncthreads();
    // G += A1 * A2 over this chunk's 256 k values; wave owns 8 k-steps.
#pragma unroll
    for (int q = 0; q < 8; ++q) {
      const int k0 = 4 * (wave * 8 + q) + 2 * lh;
      v2f a, b;
      a.x = sFa[(k0    ) * VV + lm];     // A1[v=lm, k]
      a.y = sFa[(k0 + 1) * VV + lm];
      b.x = sFb[(k0    ) * VV + lm];     // A2[k, w=lm]
      b.y = sFb[(k0 + 1) * VV + lm];
      Gacc = wmma4(a, b, Gacc);
    }
    __syncthreads();                     // sFa/sFb reused next chunk
  }

  // Reduce 8 partial G's, scale, softmax over v, add adjacency.
#pragma unroll
  for (int r = 0; r < 8; ++r)
    sG[wave][(r + 8 * lh) * VV + lm] = Gacc[r];
  __syncthreads();
  {
    const int v = tid >> 4, w = tid & 15;
    float g = 0.0f;
#pragma unroll
    for (int k = 0; k < 8; ++k) g += sG[k][v * VV + w];
    sGsum[v * VV + w] = g * (1.0f / (float)(ICN * TT));
  }
  __syncthreads();
  {
    const int v = tid >> 4, w = tid & 15;
    float mx = -3.4e38f;
    for (int vv = 0; vv < VV; ++vv) mx = fmaxf(mx, sGsum[vv * VV + w]);
    float z = 0.0f;
    for (int vv = 0; vv < VV; ++vv) z += __expf(sGsum[vv * VV + w] - mx);
    const float a  = __expf(sGsum[v * VV + w] - mx) / z;
    const int   ai = s * VV * VV + v * VV + w;
    att_out[((size_t)n * SS + s) * VV * VV + v * VV + w] =
        a + A_base[ai] + PA[ai];
  }
}

// =====================================================================
// Kernel 2: per (n, 8-t chunk):
//   Y_t = sum_s Wd[s] @ X_t @ Att[n,s];  out = Y*scale + shift + x
// =====================================================================
#define TC2    8          // t's per block
#define WDP    68         // padded Wd row stride: 16B-aligned rows for async
                          // B128; o*4 mod 64 distinct for o=0..15 (no bank dup)
#define PPITCH 17         // padded P tile row stride

__global__ __launch_bounds__(256)
void out_kernel(const float* __restrict__ x,
                const float* __restrict__ Wd, const float* __restrict__ bd,
                const float* __restrict__ gamma, const float* __restrict__ beta,
                const float* __restrict__ att, float* __restrict__ out)
{
  const int n    = blockIdx.x / (TT / TC2);
  const int t0   = (blockIdx.x % (TT / TC2)) * TC2;
  const int tid  = threadIdx.x;
  const int wave = tid >> 5;
  const int lane = tid & 31;
  const int lm   = lane & 15;
  const int lh   = lane >> 4;

  __shared__ float sX[TC2 * CC * VV];        // [t][c][v]         (32 KB)
  __shared__ float sWdS[CC * WDP];           // Wd[s], padded     (17 KB)
  __shared__ float sPw[8][16 * PPITCH];      // wave-private P    (8.5 KB)
  __shared__ float sScale[CC], sShift[CC];   //                   (0.5 KB)

  if (tid < CC) {
    const float sc   = gamma[tid] * rsqrtf(1.0f + BN_EPS);
    const float bsum = bd[tid] + bd[CC + tid] + bd[2 * CC + tid];
    sScale[tid] = sc;
    sShift[tid] = bsum * sc + beta[tid];     // (y+bd)*sc+beta = y*sc + this
  }
  // Stage the X chunk via async global->LDS: 512 rows of 16 contiguous floats.
  const float* xn = x + (size_t)n * CC * TVN;
  for (int r = tid; r < TC2 * CC; r += 256) {
    const int t = r >> 6, c = r & 63;
    async_cp64B(lds_addr32(sX + r * VV),
                xn + (size_t)c * TVN + (size_t)(t0 + t) * VV);
  }

  // Hoist Att B-operands for all 3 subsets: B[k=v, n=w].
  v2f attB[SS][4];
  for (int s = 0; s < SS; ++s)
#pragma unroll
    for (int ks = 0; ks < 4; ++ks) {
      const int v0 = 4 * ks + 2 * lh;
      const float* ap = att + ((size_t)n * SS + s) * VV * VV + v0 * VV + lm;
      attB[s][ks].x = ap[0];
      attB[s][ks].y = ap[VV];
    }

  wait_async0();                       // sX ready (per-wave ASYNCcnt)
  __syncthreads();

  const int tl = wave;                 // each wave owns one t
  float* myP = sPw[wave];
  v8f Y[4];
#pragma unroll
  for (int ot = 0; ot < 4; ++ot) Y[ot] = v8f_zero();

  for (int s = 0; s < SS; ++s) {
    __syncthreads();                   // previous sWdS fully consumed
    // Async-stage Wd[s]: 64 rows x 4 quarters, one 64B copy per thread.
    {
      const int o = tid >> 2, q = tid & 3;
      async_cp64B(lds_addr32(sWdS + o * WDP + q * 16),
                  Wd + ((size_t)s * CC + o) * CC + q * 16);
    }
    wait_async0();
    __syncthreads();

#pragma unroll
    for (int ot = 0; ot < 4; ++ot) {
      // GEMM1: P(16x16) = Wd[s] rows [ot*16..+15] @ X_t  (K = C = 64)
      v8f p = v8f_zero();
#pragma unroll
      for (int ks = 0; ks < 16; ++ks) {
        const int c0 = 4 * ks + 2 * lh;
        const int o  = ot * 16 + lm;
        v2f a, b;
        a.x = sWdS[o * WDP + c0];                 // contiguous pair
        a.y = sWdS[o * WDP + c0 + 1];
        b.x = sX[(tl * CC + c0    ) * VV + lm];
        b.y = sX[(tl * CC + c0 + 1) * VV + lm];
        p = wmma4(a, b, p);
      }
      // Transpose bounce through wave-private LDS (in-order DS per wave).
#pragma unroll
      for (int r = 0; r < 8; ++r)
        myP[(r + 8 * lh) * PPITCH + lm] = p[r];
      // GEMM2: Y[ot] += P @ Att[s]  (K = V = 16)
#pragma unroll
      for (int ks = 0; ks < 4; ++ks) {
        const int v0 = 4 * ks + 2 * lh;
        v2f a;
        a.x = myP[lm * PPITCH + v0];
        a.y = myP[lm * PPITCH + v0 + 1];
        Y[ot] = wmma4(a, attB[s][ks], Y[ot]);
      }
    }
  }

  // Epilogue: BN scale/shift (+bd) + residual from staged sX; write out.
  const int tg = t0 + tl;
  float* on = out + (size_t)n * CC * TVN + (size_t)tg * VV;
#pragma unroll
  for (int ot = 0; ot < 4; ++ot) {
#pragma unroll
    for (int r = 0; r < 8; ++r) {
      const int o = ot * 16 + r + 8 * lh;
      const float val = Y[ot][r] * sScale[o] + sShift[o]
                      + sX[(tl * CC + o) * VV + lm];
      on[(size_t)o * TVN + lm] = val;
    }
  }
}

// =====================================================================
extern "C" void kernel_launch(void* const* d_in, const int* in_sizes, int n_in,
                              void* d_out, int out_size, void* d_ws, size_t ws_size,
                              hipStream_t stream) {
  (void)in_sizes; (void)n_in; (void)out_size; (void)ws_size;
  const float* x      = (const float*)d_in[0];
  const float* A_base = (const float*)d_in[1];
  const float* PA     = (const float*)d_in[2];
  const float* Wa     = (const float*)d_in[3];
  const float* ba     = (const float*)d_in[4];
  const float* Wb     = (const float*)d_in[5];
  const float* bb     = (const float*)d_in[6];
  const float* Wd     = (const float*)d_in[7];
  const float* bd     = (const float*)d_in[8];
  const float* gamma  = (const float*)d_in[9];
  const float* beta   = (const float*)d_in[10];
  float* out = (float*)d_out;
  float* attw = (float*)d_ws;   // N*S*V*V floats = 384 KB

  attn_kernel<<<dim3(NN * SS), dim3(256), 0, stream>>>(
      x, A_base, PA, Wa, ba, Wb, bb, attw);
  out_kernel<<<dim3(NN * (TT / TC2)), dim3(256), 0, stream>>>(
      x, Wd, bd, gamma, beta, attw, out);
}